// GeometrySimilarity_38070590112397
// MI455X (gfx1250) — compile-verified
//
#include <hip/hip_runtime.h>

#define EPS 1e-5f
#define TDIM 4096
#define DDIM 64
#define KDIM 32
#define NEXT 48          // 32 centroids + 4 freq rows + 4 phase rows + 8 pad
#define TILE_T 128
#define XS 68            // padded LDS row stride (floats) -> conflict-free ds_load_b64
#define CMS 68

typedef float v2f __attribute__((ext_vector_type(2)));
typedef float v4f __attribute__((ext_vector_type(4)));
typedef float v8f __attribute__((ext_vector_type(8)));

__device__ __forceinline__ float branch_mix(
    float dot, float x2, float sx, float c2, float sc,
    const float* tf, const float* co, const float* si,
    const float* cf, const float* cc, const float* cs,
    float c_euc, float c_hyp, float c_frc, float c_osc,
    float alpha, float opw)
{
    float d2 = fmaxf(x2 + c2 - 2.0f * dot, 0.0f);
    float s = -c_euc * d2;
    // fractal: (d2+eps)^alpha
    s -= c_frc * __expf(alpha * __logf(d2 + EPS));
    // hyperbolic: projection is a per-row scale -> reuse the same dot product
    float xb2 = sx * sx * x2, cb2 = sc * sc * c2;
    float d2b = fmaxf(xb2 + cb2 - 2.0f * sx * sc * dot, 0.0f);
    float denom = (1.0f - xb2) * (1.0f - cb2) + EPS;
    float arg = fmaxf(1.0f + 2.0f * d2b / denom, 1.0f + 1e-7f);
    float ach = __logf(arg + sqrtf(arg * arg - 1.0f));
    s -= c_hyp * ach;
    // oscillator: freq L2 + phase mismatch via cos(a-b)=ca*cb+sa*sb
    float fd2 = 0.0f, pc = 0.0f;
#pragma unroll
    for (int p = 0; p < 4; ++p) {
        float df = tf[p] - cf[p];
        fd2 += df * df;
        pc += co[p] * cc[p] + si[p] * cs[p];
    }
    float pm = 1.0f - 0.25f * pc;
    s -= c_osc * (fd2 + opw * pm);
    return s;
}

__launch_bounds__(256)
__global__ void geom_sim_kernel(
    const float* __restrict__ x, const float* __restrict__ centroids,
    const float* __restrict__ Wf, const float* __restrict__ bf,
    const float* __restrict__ Wp, const float* __restrict__ bp,
    const float* __restrict__ cfreq, const float* __restrict__ cphase,
    const float* __restrict__ opw_p, const float* __restrict__ alpha_p,
    const float* __restrict__ tau_p, const float* __restrict__ gate_p,
    float* __restrict__ out)
{
    __shared__ float lds_x[TILE_T * XS];     // 34816 B, async-staged x tile
    __shared__ float lds_cm[NEXT * CMS];     // 13056 B, combined B matrix [n][d]
    __shared__ float lds_c2[KDIM], lds_sc[KDIM];
    __shared__ float lds_cf[KDIM * 4], lds_cc[KDIM * 4], lds_cs[KDIM * 4];
    __shared__ float lds_osc[8][16 * 8];     // per-wave raw freq/phase scratch
    __shared__ float lds_tok[8][16 * 16];    // per-wave token stats

    const int tid = threadIdx.x;
    const int wgid = blockIdx.x;
    const int nchunk = TDIM / TILE_T;        // 32
    const int tc = wgid % nchunk;
    const int bh = wgid / nchunk;            // b*8 + h
    const int h = bh & 7;

    const float* xbase = x + ((size_t)bh * TDIM + (size_t)tc * TILE_T) * DDIM;
    float* obase = out + ((size_t)bh * TDIM + (size_t)tc * TILE_T) * KDIM;

    // ---------- Phase 1: async-stage x tile; build combined matrix ----------
#pragma unroll
    for (int it = 0; it < (TILE_T * DDIM) / (256 * 4); ++it) {
        int g = (it * 256 + tid) * 4;
        int row = g >> 6, col = g & 63;
        unsigned laddr = (unsigned)(uintptr_t)&lds_x[row * XS + col];
        const float* gaddr = xbase + g;
        asm volatile("global_load_async_to_lds_b128 %0, %1, off"
                     :: "v"(laddr), "v"(gaddr) : "memory");
    }
    // rows [0,32)=centroids[h], [32,36)=Wf, [36,40)=Wp, [40,48)=0
    for (int i = tid; i < NEXT * DDIM; i += 256) {
        int n = i >> 6, d = i & 63;
        float v = 0.0f;
        if (n < KDIM)                v = centroids[((size_t)h * KDIM + n) * DDIM + d];
        else if (n < KDIM + 4)       v = Wf[(n - KDIM) * DDIM + d];
        else if (n < KDIM + 8)       v = Wp[(n - KDIM - 4) * DDIM + d];
        lds_cm[n * CMS + d] = v;
    }

    // uniform scalars: softmax(gate)/tau folded into one coefficient per branch
    float tau0 = fmaxf(tau_p[0], EPS), tau1 = fmaxf(tau_p[1], EPS);
    float tau2 = fmaxf(tau_p[2], EPS), tau3 = fmaxf(tau_p[3], EPS);
    float g0 = gate_p[0], g1 = gate_p[1], g2 = gate_p[2], g3 = gate_p[3];
    float gm = fmaxf(fmaxf(g0, g1), fmaxf(g2, g3));
    float e0 = __expf(g0 - gm), e1 = __expf(g1 - gm);
    float e2 = __expf(g2 - gm), e3 = __expf(g3 - gm);
    float inv_es = 1.0f / (e0 + e1 + e2 + e3);
    float c_euc = e0 * inv_es / tau0, c_hyp = e1 * inv_es / tau1;
    float c_frc = e2 * inv_es / tau2, c_osc = e3 * inv_es / tau3;
    float alpha = alpha_p[0], opw = opw_p[0];
    float bfv[4], bpv[4];
#pragma unroll
    for (int p = 0; p < 4; ++p) { bfv[p] = bf[p]; bpv[p] = bp[p]; }

    asm volatile("s_wait_asynccnt 0x0" ::: "memory");
    __syncthreads();

    // ---------- Phase 2: per-centroid stats ----------
    if (tid < KDIM) {
        int k = tid;
        float c2 = 0.0f;
        for (int d0 = 0; d0 < DDIM; d0 += 4) {
            v4f c = *(const v4f*)&lds_cm[k * CMS + d0];
            c2 += c.x * c.x + c.y * c.y + c.z * c.z + c.w * c.w;
        }
        lds_c2[k] = c2;
        lds_sc[k] = fminf(1.0f, (1.0f - EPS) / (sqrtf(c2) + EPS));
#pragma unroll
        for (int p = 0; p < 4; ++p) {
            lds_cf[k * 4 + p] = cfreq[k * 4 + p];
            float ph = cphase[k * 4 + p];
            lds_cc[k * 4 + p] = __cosf(ph);
            lds_cs[k * 4 + p] = __sinf(ph);
        }
    }
    __syncthreads();

    // ---------- Phase 3: f32 WMMA GEMM, 16 tokens x 48 cols per wave ----------
    const int wv = tid >> 5, ln = tid & 31;
    const int half = ln >> 4, n16 = ln & 15;

    v8f acc0 = {0,0,0,0,0,0,0,0};
    v8f acc1 = {0,0,0,0,0,0,0,0};
    v8f acc2 = {0,0,0,0,0,0,0,0};
    // A frag (16x4 f32): lane L comp j -> x[M=L%16][K=ks+2*(L/16)+j]
    const float* arow = &lds_x[(wv * 16 + n16) * XS + 2 * half];
    // B frag (4x16 f32): lane L comp j -> CM[n][ks+2*(L/16)+j]
    const float* b0 = &lds_cm[(0 * 16 + n16) * CMS + 2 * half];
    const float* b1 = &lds_cm[(1 * 16 + n16) * CMS + 2 * half];
    const float* b2 = &lds_cm[(2 * 16 + n16) * CMS + 2 * half];
#pragma unroll
    for (int ks = 0; ks < DDIM; ks += 4) {
        v2f a   = *(const v2f*)(arow + ks);
        v2f bb0 = *(const v2f*)(b0 + ks);
        v2f bb1 = *(const v2f*)(b1 + ks);
        v2f bb2 = *(const v2f*)(b2 + ks);
        acc0 = __builtin_amdgcn_wmma_f32_16x16x4_f32(false, a, false, bb0, (short)0, acc0, false, false);
        acc1 = __builtin_amdgcn_wmma_f32_16x16x4_f32(false, a, false, bb1, (short)0, acc1, false, false);
        acc2 = __builtin_amdgcn_wmma_f32_16x16x4_f32(false, a, false, bb2, (short)0, acc2, false, false);
    }

    // ---------- Phase 4: per-wave epilogue ----------
    // spill raw freq(0..3)/phase(4..7) columns to LDS scratch (cross-lane)
    if (n16 < 8) {
#pragma unroll
        for (int g = 0; g < 8; ++g)
            lds_osc[wv][(g + 8 * half) * 8 + n16] = acc2[g];
    }
    asm volatile("s_wait_dscnt 0x0" ::: "memory");

    // lanes 0..15: per-token stats (norm, ball scale, freq+bias, cos/sin phase)
    if (ln < 16) {
        int m = ln;
        int r = wv * 16 + m;
        float x2 = 0.0f;
        for (int d0 = 0; d0 < DDIM; d0 += 4) {
            v4f xv = *(const v4f*)&lds_x[r * XS + d0];
            x2 += xv.x * xv.x + xv.y * xv.y + xv.z * xv.z + xv.w * xv.w;
        }
        float sx = fminf(1.0f, (1.0f - EPS) / (sqrtf(x2) + EPS));
        float* tk = &lds_tok[wv][m * 16];
        tk[0] = x2; tk[1] = sx;
#pragma unroll
        for (int p = 0; p < 4; ++p) {
            float tf = lds_osc[wv][m * 8 + p] + bfv[p];
            float tp = lds_osc[wv][m * 8 + 4 + p] + bpv[p];
            tk[2 + p]  = tf;
            tk[6 + p]  = __cosf(tp);
            tk[10 + p] = __sinf(tp);
        }
    }
    asm volatile("s_wait_dscnt 0x0" ::: "memory");

    // this lane's two output columns
    const int k0 = n16, k1 = n16 + 16;
    float c2a = lds_c2[k0], sca = lds_sc[k0];
    float c2b = lds_c2[k1], scb = lds_sc[k1];
    float cfa[4], cca[4], csa[4], cfb[4], ccb[4], csb[4];
#pragma unroll
    for (int p = 0; p < 4; ++p) {
        cfa[p] = lds_cf[k0 * 4 + p]; cca[p] = lds_cc[k0 * 4 + p]; csa[p] = lds_cs[k0 * 4 + p];
        cfb[p] = lds_cf[k1 * 4 + p]; ccb[p] = lds_cc[k1 * 4 + p]; csb[p] = lds_cs[k1 * 4 + p];
    }

#pragma unroll
    for (int g = 0; g < 8; ++g) {
        int m = g + 8 * half;
        const float* tk = &lds_tok[wv][m * 16];
        float x2 = tk[0], sx = tk[1];
        float tfv[4], cov[4], siv[4];
#pragma unroll
        for (int p = 0; p < 4; ++p) { tfv[p] = tk[2 + p]; cov[p] = tk[6 + p]; siv[p] = tk[10 + p]; }
        float* orow = obase + (size_t)(wv * 16 + m) * KDIM;
        orow[k0] = branch_mix(acc0[g], x2, sx, c2a, sca, tfv, cov, siv,
                              cfa, cca, csa, c_euc, c_hyp, c_frc, c_osc, alpha, opw);
        orow[k1] = branch_mix(acc1[g], x2, sx, c2b, scb, tfv, cov, siv,
                              cfb, ccb, csb, c_euc, c_hyp, c_frc, c_osc, alpha, opw);
    }
}

extern "C" void kernel_launch(void* const* d_in, const int* in_sizes, int n_in,
                              void* d_out, int out_size, void* d_ws, size_t ws_size,
                              hipStream_t stream) {
    (void)in_sizes; (void)n_in; (void)out_size; (void)d_ws; (void)ws_size;
    const float* x         = (const float*)d_in[0];
    const float* centroids = (const float*)d_in[1];
    const float* Wf        = (const float*)d_in[2];
    const float* bf        = (const float*)d_in[3];
    const float* Wp        = (const float*)d_in[4];
    const float* bp        = (const float*)d_in[5];
    const float* cfreq     = (const float*)d_in[6];
    const float* cphase    = (const float*)d_in[7];
    const float* opw       = (const float*)d_in[8];
    const float* alpha     = (const float*)d_in[9];
    const float* tau       = (const float*)d_in[10];
    const float* gate      = (const float*)d_in[11];

    dim3 grid(64 * (TDIM / TILE_T));   // (B*H) * t-chunks = 2048
    dim3 block(256);                   // 8 wave32 per workgroup
    geom_sim_kernel<<<grid, block, 0, stream>>>(
        x, centroids, Wf, bf, Wp, bp, cfreq, cphase, opw, alpha, tau, gate,
        (float*)d_out);
}